// GCN_1382979469383
// MI455X (gfx1250) — compile-verified
//
#include <hip/hip_runtime.h>
#include <hip/hip_bf16.h>
#include <math.h>

typedef __attribute__((ext_vector_type(2))) float v2f;
typedef __attribute__((ext_vector_type(8))) float v8f;

// ---------------------------------------------------------------- degree ----
__global__ void k_deg_init(float* __restrict__ deg, int N) {
    int i = blockIdx.x * blockDim.x + threadIdx.x;
    if (i < N) deg[i] = 1.0f;                       // self-loop contributes 1
}

__global__ void k_deg_count(const long long* __restrict__ ei,
                            float* __restrict__ deg, int E) {
    int e = blockIdx.x * blockDim.x + threadIdx.x;
    if (e < E) {
        int d = (int)ei[(size_t)E + e];             // dst row of edge_index
        atomicAdd(&deg[d], 1.0f);
    }
}

__global__ void k_deg_fin(float* __restrict__ deg, int N) {
    int i = blockIdx.x * blockDim.x + threadIdx.x;
    if (i < N) deg[i] = rsqrtf(deg[i]);             // deg >= 1 always
}

// ------------------------------------------------- dense: out = A @ W (64) --
// A: N x 64, W: 64 x 64, out: N x 64.  FP32 WMMA 16x16x4, K-loop of 16.
// Block = 256 threads = 8 waves = 2 row-tiles x 4 col-tiles = 32 rows/block.
__global__ void k_gemm64(const float* __restrict__ A, const float* __restrict__ W,
                         float* __restrict__ out, int N) {
    const int lane = threadIdx.x & 31;
    const int wave = threadIdx.x >> 5;
    const int mt   = wave >> 2;            // 0..1
    const int nt   = wave & 3;             // 0..3
    const int row0 = blockIdx.x * 32 + mt * 16;
    const int col0 = nt * 16;
    const int l16  = lane & 15;
    const bool hi  = lane >= 16;
    const int khi  = hi ? 2 : 0;

    // Preload all B fragments (W is uniform across blocks, lives in cache).
    v2f bfrag[16];
#pragma unroll
    for (int k = 0; k < 16; ++k) {
        const float* wp = W + (size_t)(k * 4 + khi) * 64 + col0 + l16;
        bfrag[k].x = wp[0];                // K = 4k+khi
        bfrag[k].y = wp[64];               // K = 4k+khi+1
    }

    int arow = row0 + l16;
    if (arow > N - 1) arow = N - 1;        // clamp (keeps EXEC all-ones)
    const float* ap = A + (size_t)arow * 64 + khi;

    v8f c = {};
#pragma unroll
    for (int k = 0; k < 16; ++k) {
        v2f a;
        a.x = ap[k * 4 + 0];
        a.y = ap[k * 4 + 1];
        c = __builtin_amdgcn_wmma_f32_16x16x4_f32(false, a, false, bfrag[k],
                                                  (short)0, c, false, false);
    }

#pragma unroll
    for (int v = 0; v < 8; ++v) {
        int r = row0 + v + (hi ? 8 : 0);
        if (r < N) out[(size_t)r * 64 + col0 + l16] = c[v];
    }
}

// ------------------------------------------------- fc: out = A @ Wfc (16) ---
// A: N x 64, W: 64 x 16, out(logits): N x 16.  One 16x16 tile per wave.
__global__ void k_fc16(const float* __restrict__ A, const float* __restrict__ W,
                       float* __restrict__ out, int N) {
    const int lane = threadIdx.x & 31;
    const int wave = threadIdx.x >> 5;
    const int row0 = (blockIdx.x * 8 + wave) * 16;
    const int l16  = lane & 15;
    const bool hi  = lane >= 16;
    const int khi  = hi ? 2 : 0;

    v2f bfrag[16];
#pragma unroll
    for (int k = 0; k < 16; ++k) {
        const float* wp = W + (size_t)(k * 4 + khi) * 16 + l16;
        bfrag[k].x = wp[0];
        bfrag[k].y = wp[16];
    }

    int arow = row0 + l16;
    if (arow > N - 1) arow = N - 1;
    const float* ap = A + (size_t)arow * 64 + khi;

    v8f c = {};
#pragma unroll
    for (int k = 0; k < 16; ++k) {
        v2f a;
        a.x = ap[k * 4 + 0];
        a.y = ap[k * 4 + 1];
        c = __builtin_amdgcn_wmma_f32_16x16x4_f32(false, a, false, bfrag[k],
                                                  (short)0, c, false, false);
    }

#pragma unroll
    for (int v = 0; v < 8; ++v) {
        int r = row0 + v + (hi ? 8 : 0);
        if (r < N) out[(size_t)r * 16 + l16] = c[v];
    }
}

// --------------------------------------------------------- aggregation -----
// agg[i,:] = dinv[i]^2 * h[i,:] + b[:]   (self-loop + bias)
__global__ void k_agg_init(const float* __restrict__ h, const float* __restrict__ dinv,
                           const float* __restrict__ b, float* __restrict__ agg, int NF) {
    int i = blockIdx.x * blockDim.x + threadIdx.x;
    if (i < NF) {
        int row = i >> 6;
        int f   = i & 63;
        float di = dinv[row];
        agg[i] = di * di * h[i] + b[f];
    }
}

// agg[dst,:] += dinv[src]*dinv[dst] * h[src,:]   (wave per edge, 2 feats/lane)
__global__ void k_agg_edges(const long long* __restrict__ ei, const float* __restrict__ dinv,
                            const float* __restrict__ h, float* __restrict__ agg, int E) {
    long long gid = (long long)blockIdx.x * blockDim.x + threadIdx.x;
    int e    = (int)(gid >> 5);
    int lane = (int)(gid & 31);
    if (e >= E) return;
    int s = (int)ei[e];
    int d = (int)ei[(size_t)E + e];
    float w = dinv[s] * dinv[d];
    const float* hp = h + (size_t)s * 64;
    float* ap = agg + (size_t)d * 64;
    atomicAdd(&ap[lane],      w * hp[lane]);
    atomicAdd(&ap[lane + 32], w * hp[lane + 32]);
}

__global__ void k_relu(float* __restrict__ x, int NF) {
    int i = blockIdx.x * blockDim.x + threadIdx.x;
    if (i < NF) x[i] = fmaxf(x[i], 0.0f);
}

// ----------------------------------------------------------- softmax -------
__global__ void k_softmax16(const float* __restrict__ logits, const float* __restrict__ bfc,
                            float* __restrict__ out, int N) {
    int row = blockIdx.x * blockDim.x + threadIdx.x;
    if (row >= N) return;
    float v[16];
    float m = -3.402823466e38f;
#pragma unroll
    for (int i = 0; i < 16; ++i) {
        v[i] = logits[(size_t)row * 16 + i] + bfc[i];
        m = fmaxf(m, v[i]);
    }
    float s = 0.0f;
#pragma unroll
    for (int i = 0; i < 16; ++i) {
        v[i] = expf(v[i] - m);
        s += v[i];
    }
    float inv = 1.0f / s;
#pragma unroll
    for (int i = 0; i < 16; ++i)
        out[(size_t)row * 16 + i] = v[i] * inv;
}

// ----------------------------------------------------------------- launch --
extern "C" void kernel_launch(void* const* d_in, const int* in_sizes, int n_in,
                              void* d_out, int out_size, void* d_ws, size_t ws_size,
                              hipStream_t stream) {
    const float*     x   = (const float*)d_in[0];
    const long long* ei  = (const long long*)d_in[1];
    const float*     W1  = (const float*)d_in[2];
    const float*     b1  = (const float*)d_in[3];
    const float*     W2  = (const float*)d_in[4];
    const float*     b2  = (const float*)d_in[5];
    const float*     Wfc = (const float*)d_in[6];
    const float*     bfc = (const float*)d_in[7];
    float*           outp = (float*)d_out;

    const int N = in_sizes[0] / 64;     // 100000
    const int E = in_sizes[1] / 2;      // 1600000
    const int NF = N * 64;

    float* dinv = (float*)d_ws;
    float* bufA = dinv + N;             // N*64  (dense transform output)
    float* bufB = bufA + (size_t)N * 64;// N*64  (aggregation output)

    const int T = 256;
    const int nBlk  = (N + T - 1) / T;
    const int eBlk  = (E + T - 1) / T;
    const int nfBlk = (NF + T - 1) / T;
    const int gBlk  = (N + 31) / 32;                         // gemm64 tiles
    const long long eThreads = (long long)E * 32;
    const int eaBlk = (int)((eThreads + T - 1) / T);         // wave-per-edge
    const int fcBlk = (N + 127) / 128;                       // 8 tiles/block

    // normalization
    k_deg_init <<<nBlk,  T, 0, stream>>>(dinv, N);
    k_deg_count<<<eBlk,  T, 0, stream>>>(ei, dinv, E);
    k_deg_fin  <<<nBlk,  T, 0, stream>>>(dinv, N);

    // layer 1: h = relu(Agg(x @ W1) + b1)
    k_gemm64   <<<gBlk, 256, 0, stream>>>(x, W1, bufA, N);
    k_agg_init <<<nfBlk, T, 0, stream>>>(bufA, dinv, b1, bufB, NF);
    k_agg_edges<<<eaBlk, T, 0, stream>>>(ei, dinv, bufA, bufB, E);
    k_relu     <<<nfBlk, T, 0, stream>>>(bufB, NF);

    // layer 2: h = relu(Agg(h @ W2) + b2)
    k_gemm64   <<<gBlk, 256, 0, stream>>>(bufB, W2, bufA, N);
    k_agg_init <<<nfBlk, T, 0, stream>>>(bufA, dinv, b2, bufB, NF);
    k_agg_edges<<<eaBlk, T, 0, stream>>>(ei, dinv, bufA, bufB, E);
    k_relu     <<<nfBlk, T, 0, stream>>>(bufB, NF);

    // fc + softmax (logits reuse bufA)
    k_fc16     <<<fcBlk, 256, 0, stream>>>(bufB, Wfc, bufA, N);
    k_softmax16<<<nBlk,  T, 0, stream>>>(bufA, bfc, outp, N);
}